// TopKSparseRouter_19267223289857
// MI455X (gfx1250) — compile-verified
//
#include <hip/hip_runtime.h>

typedef float v2f __attribute__((ext_vector_type(2)));
typedef float v8f __attribute__((ext_vector_type(8)));

#define B_    8
#define H_    16
#define Q_    64
#define D_    64
#define C_    8192
#define KMAX_ 64

// ---------------------------------------------------------------------------
// Kernel 1: per (b, c) compute max over (h, q) of  q[b,h,q,:] . k[b,c,h,:]
// One wave (32 threads) handles a 32-wide c-tile (2 WMMA tiles) for one batch.
// Uses V_WMMA_F32_16X16X4_F32: exact f32 math; memory-bound on the key stream.
// ---------------------------------------------------------------------------
__global__ __launch_bounds__(32) void scores_max_kernel(
    const float* __restrict__ qry, const float* __restrict__ keys,
    float* __restrict__ colmax_out)
{
    const int lane = threadIdx.x;        // 0..31
    const int nn   = lane & 15;          // row (A) / col (B) within tile
    const int half = lane >> 4;          // 0: k=0,1   1: k=2,3
    const int b    = blockIdx.y;
    const int c0   = blockIdx.x * 32;    // two 16-wide c-tiles

    float cm0 = -__builtin_inff();
    float cm1 = -__builtin_inff();

    // queries: ((b*H + h)*Q + row)*D + d
    // keys:    ((b*C + c)*H + h)*D + d
    const size_t qBase = (size_t)b * H_ * Q_ * D_;
    const size_t kBase = ((size_t)b * C_ + (size_t)(c0 + nn)) * H_ * D_;

    for (int h = 0; h < H_; ++h) {
        v8f acc[8];
        #pragma unroll
        for (int i = 0; i < 8; ++i) acc[i] = (v8f){};

        const float* aPtr  = qry  + qBase + ((size_t)h * Q_ + nn) * D_ + half * 2;
        const float* bPtr0 = keys + kBase + (size_t)h * D_ + half * 2;
        const float* bPtr1 = bPtr0 + (size_t)16 * H_ * D_;   // +16 columns of c

        #pragma unroll
        for (int ks = 0; ks < D_ / 4; ++ks) {
            const int d = ks * 4;
            v2f a0 = *(const v2f*)(aPtr + 0 * 16 * D_ + d);
            v2f a1 = *(const v2f*)(aPtr + 1 * 16 * D_ + d);
            v2f a2 = *(const v2f*)(aPtr + 2 * 16 * D_ + d);
            v2f a3 = *(const v2f*)(aPtr + 3 * 16 * D_ + d);
            v2f b0 = *(const v2f*)(bPtr0 + d);
            v2f b1 = *(const v2f*)(bPtr1 + d);

            acc[0] = __builtin_amdgcn_wmma_f32_16x16x4_f32(false, a0, false, b0, (short)0, acc[0], false, false);
            acc[1] = __builtin_amdgcn_wmma_f32_16x16x4_f32(false, a0, false, b1, (short)0, acc[1], false, false);
            acc[2] = __builtin_amdgcn_wmma_f32_16x16x4_f32(false, a1, false, b0, (short)0, acc[2], false, false);
            acc[3] = __builtin_amdgcn_wmma_f32_16x16x4_f32(false, a1, false, b1, (short)0, acc[3], false, false);
            acc[4] = __builtin_amdgcn_wmma_f32_16x16x4_f32(false, a2, false, b0, (short)0, acc[4], false, false);
            acc[5] = __builtin_amdgcn_wmma_f32_16x16x4_f32(false, a2, false, b1, (short)0, acc[5], false, false);
            acc[6] = __builtin_amdgcn_wmma_f32_16x16x4_f32(false, a3, false, b0, (short)0, acc[6], false, false);
            acc[7] = __builtin_amdgcn_wmma_f32_16x16x4_f32(false, a3, false, b1, (short)0, acc[7], false, false);
        }

        // C/D layout: VGPR v, lanes 0-15 -> row v, col=lane; lanes 16-31 -> row v+8.
        // Per-lane max over the 8 accumulator rows; per-lane column is fixed.
        #pragma unroll
        for (int qt = 0; qt < 4; ++qt) {
            #pragma unroll
            for (int v = 0; v < 8; ++v) {
                cm0 = fmaxf(cm0, acc[qt * 2 + 0][v]);
                cm1 = fmaxf(cm1, acc[qt * 2 + 1][v]);
            }
        }
    }

    // merge row-halves (lanes n and n+16 hold the same column, different rows)
    cm0 = fmaxf(cm0, __shfl_xor(cm0, 16, 32));
    cm1 = fmaxf(cm1, __shfl_xor(cm1, 16, 32));

    const float scale = 0.125f;   // rsqrt(64); max commutes with positive scale
    float* out = colmax_out + (size_t)b * C_ + c0;
    if (half == 0) out[nn]      = cm0 * scale;
    else           out[16 + nn] = cm1 * scale;
}

// ---------------------------------------------------------------------------
// Kernel 2: per-batch top-k over C=8192 scores. One block per batch,
// k sequential argmax passes; ties break to the smaller index (jax top_k).
// ---------------------------------------------------------------------------
__global__ __launch_bounds__(256) void topk_kernel(
    const float* __restrict__ scores, const int* __restrict__ topk_ptr,
    void* __restrict__ d_out)
{
    __shared__ float sv[256];
    __shared__ int   si[256];
    __shared__ int   selI[KMAX_];
    __shared__ float selV[KMAX_];

    const int tid = threadIdx.x;
    const int b   = blockIdx.x;
    const float* row = scores + (size_t)b * C_;

    int kk = topk_ptr[0];
    if (kk > KMAX_) kk = KMAX_;
    if (kk > C_)    kk = C_;

    for (int j = 0; j < kk; ++j) {
        float bv = -__builtin_inff();
        int   bi = 0x7fffffff;
        for (int i = tid; i < C_; i += 256) {
            bool taken = false;
            for (int t = 0; t < j; ++t) taken |= (selI[t] == i);
            float v = row[i];
            if (!taken && (v > bv || (v == bv && i < bi))) { bv = v; bi = i; }
        }
        sv[tid] = bv; si[tid] = bi;
        __syncthreads();
        #pragma unroll
        for (int s = 128; s > 0; s >>= 1) {
            if (tid < s) {
                float v2 = sv[tid + s]; int i2 = si[tid + s];
                if (v2 > sv[tid] || (v2 == sv[tid] && i2 < si[tid])) {
                    sv[tid] = v2; si[tid] = i2;
                }
            }
            __syncthreads();
        }
        if (tid == 0) { selI[j] = si[0]; selV[j] = sv[0]; }
        __syncthreads();
    }

    // d_out layout: B*kk int32 indices, then B*kk f32 scores
    int*   outIdx = (int*)d_out;
    float* outVal = (float*)d_out + (size_t)B_ * kk;
    if (tid < kk) {
        outIdx[b * kk + tid] = selI[tid];
        outVal[b * kk + tid] = selV[tid];
    }
}

extern "C" void kernel_launch(void* const* d_in, const int* in_sizes, int n_in,
                              void* d_out, int out_size, void* d_ws, size_t ws_size,
                              hipStream_t stream) {
    const float* qry  = (const float*)d_in[0];   // (B,H,Q,D) f32
    const float* keys = (const float*)d_in[1];   // (B,C,H,D) f32
    const int*   topk = (const int*)d_in[2];     // scalar int

    float* ws_scores = (float*)d_ws;             // (B,C) f32, 256 KB

    dim3 grid1(C_ / 32, B_);
    scores_max_kernel<<<grid1, 32, 0, stream>>>(qry, keys, ws_scores);
    topk_kernel<<<B_, 256, 0, stream>>>(ws_scores, topk, d_out);
}